// TGVDenoiser_67413806678461
// MI455X (gfx1250) — compile-verified
//
#include <hip/hip_runtime.h>
#include <math.h>

#define HDIM 1024
#define WDIM 1024
#define NB   4
#define TH   16
#define TW   64
#define NTHREADS 256

#define TAU_F   0.01f
#define RHO_F   1.99f
#define SIGMA_F (1.0f / 0.01f / 72.0f)

// LDS region extents (tile-relative origins noted)
#define U2R (TH + 5)   // u2 rows,  origin h0-3
#define U2C (TW + 5)   // u2 cols,  origin w0-3
#define TMR (TH + 3)   // tmp rows, origin h0-2
#define TMC (TW + 3)   // tmp cols, origin w0-2
#define XBR (TH + 2)   // xb rows,  origin h0-1
#define XBC (TW + 2)   // xb cols,  origin w0-1
#define RBR (TH + 2)   // rb rows,  origin h0-1
#define RBC (TW + 1)   // rb cols,  origin w0-1

__global__ __launch_bounds__(NTHREADS)
void tgv_pdhg_step(const float* __restrict__ x2_in,
                   const float* __restrict__ r2_in,
                   const float* __restrict__ u2_in,
                   const float* __restrict__ yv,
                   float* __restrict__ x2_out,
                   float* __restrict__ r2_out,
                   float* __restrict__ u2_out,
                   const float* __restrict__ ths_p,
                   int first, int last)
{
  __shared__ alignas(16) float u2s[U2R * U2C * 4];  // staged u2 tile (AoS, 16B cells)
  __shared__ float tms[TMR * TMC * 2];              // tmp = TAU * eps2_adj(u2)
  __shared__ float xbs[XBR * XBC];                  // xb = 2x - x2_old
  __shared__ float rbs[RBR * RBC * 2];              // rb = 2r - r2_old

  const int tid = threadIdx.x;
  const int b  = blockIdx.z;
  const int h0 = blockIdx.y * TH;
  const int w0 = blockIdx.x * TW;
  const int bbase = b * (HDIM * WDIM);

  const float ths  = ths_p[0];
  const float lam1 = 0.10f * ths;
  const float lam2 = 0.15f * ths;
  const float inv_tau_lam1 = 1.0f / (TAU_F * lam1);
  const float inv_lam2     = 1.0f / lam2;
  const float inv_1p_tau   = 1.0f / (1.0f + TAU_F);

  // ---- stage 0: async global->LDS staging of the u2 tile (+halo). CDNA5 path.
  if (!first) {
    for (int t = tid; t < U2R * U2C; t += NTHREADS) {
      const int ri = t / U2C;
      const int rj = t - ri * U2C;
      const int gi = h0 - 3 + ri;
      const int gj = w0 - 3 + rj;
      if (gi >= 0 && gi < HDIM && gj >= 0 && gj < WDIM) {
        const float* src = u2_in + (size_t)(bbase + gi * WDIM + gj) * 4;
        const unsigned loff = (unsigned)(unsigned long long)&u2s[t * 4];
        asm volatile("global_load_async_to_lds_b128 %0, %1, off"
                     :: "v"(loff), "v"(src) : "memory");
      }
      // out-of-image halo cells are never read (all consumers are image-bounds guarded)
    }
    asm volatile("s_wait_asynccnt 0x0" ::: "memory");
  }
  __syncthreads();

  // ---- stage 1: tmp = TAU * epsilon2_adjoint(u2) over [-2,TH] x [-2,TW]
  for (int t = tid; t < TMR * TMC; t += NTHREADS) {
    const int ri = t / TMC;
    const int rj = t - ri * TMC;
    const int gi = h0 - 2 + ri;
    const int gj = w0 - 2 + rj;
    float t0 = 0.f, t1 = 0.f;
    if (!first && gi >= 0 && gi < HDIM && gj >= 0 && gj < WDIM) {
      const int ur = ri + 1, uc = rj + 1;  // u2 LDS coords
      #define UC(c, rr, cc) u2s[(((rr) * U2C) + (cc)) * 4 + (c)]
      const float a0 = UC(0, ur, uc);
      const float a1 = (gi < HDIM - 1) ? UC(0, ur + 1, uc) : 0.f;
      const float a2 = (gj > 0)        ? UC(1, ur, uc)     : 0.f;
      const float a3 = (gj < WDIM - 1) ? UC(1, ur, uc + 1) : 0.f;
      const float a4 = UC(2, ur, uc);
      const float a5 = (gj < WDIM - 1) ? UC(2, ur, uc + 1) : 0.f;
      const float a6 = (gi > 0)        ? UC(3, ur - 1, uc) : 0.f;
      const float a7 = (gi < HDIM - 1) ? UC(3, ur, uc)     : 0.f;
      t0 = TAU_F * (a0 - a1 + a2 - a3);
      t1 = TAU_F * (a4 - a5 + a6 - a7);
    }
    tms[t * 2 + 0] = t0;
    tms[t * 2 + 1] = t1;
  }
  __syncthreads();

  // ---- stage 2a: x-update over [-1,TH] x [-1,TW]; write relaxed x2 on interior
  for (int t = tid; t < XBR * XBC; t += NTHREADS) {
    const int ri = t / XBC;
    const int rj = t - ri * XBC;
    const int gi = h0 - 1 + ri;
    const int gj = w0 - 1 + rj;
    float xb = 0.f;
    if (gi >= 0 && gi < HDIM && gj >= 0 && gj < WDIM) {
      const int tr = ri + 1, tc = rj + 1;  // tmp LDS coords
      const float m0 = (gi > 0)        ? tms[(((tr - 1) * TMC) + tc) * 2 + 0] : 0.f;
      const float m1 = (gi < HDIM - 1) ? tms[((tr * TMC) + tc) * 2 + 0]       : 0.f;
      const float m2 = (gj > 0)        ? tms[((tr * TMC) + (tc - 1)) * 2 + 1] : 0.f;
      const float m3 = (gj < WDIM - 1) ? tms[((tr * TMC) + tc) * 2 + 1]       : 0.f;
      const float adj = m0 - m1 + m2 - m3;           // nabla2_adjoint(tmp)
      const int gidx = bbase + gi * WDIM + gj;
      const float x2v = x2_in[gidx];                 // == y on first iteration
      const float yy  = yv[gidx];
      const float x   = (x2v - adj + TAU_F * yy) * inv_1p_tau;
      xb = 2.f * x - x2v;
      if (ri >= 1 && ri <= TH && rj >= 1 && rj <= TW) {
        x2_out[gidx] = x2v + RHO_F * (x - x2v);
      }
    }
    xbs[t] = xb;
  }

  if (!last) {
    // ---- stage 2b: r-update over [-1,TH] x [-1,TW-1]; write relaxed r2 on interior
    for (int t = tid; t < RBR * RBC; t += NTHREADS) {
      const int ri = t / RBC;
      const int rj = t - ri * RBC;
      const int gi = h0 - 1 + ri;
      const int gj = w0 - 1 + rj;
      float rb0 = 0.f, rb1 = 0.f;
      if (gi >= 0 && gi < HDIM && gj >= 0 && gj < WDIM) {
        const int tr = ri + 1, tc = rj + 1;
        const float t0 = tms[((tr * TMC) + tc) * 2 + 0];
        const float t1 = tms[((tr * TMC) + tc) * 2 + 1];
        const int gidx = bbase + gi * WDIM + gj;
        float r20 = 0.f, r21 = 0.f;
        if (!first) {
          const float2 rv = *(const float2*)(r2_in + (size_t)gidx * 2);
          r20 = rv.x; r21 = rv.y;
        }
        const float rr0 = r20 + t0;
        const float rr1 = r21 + t1;
        const float left = sqrtf(rr0 * rr0 + rr1 * rr1) * inv_tau_lam1;
        const float m = fmaxf(left, 1.0f);
        const float s = 1.0f - 1.0f / m;             // r = rr - rr/m
        const float r0 = rr0 * s;
        const float r1 = rr1 * s;
        rb0 = 2.f * r0 - r20;
        rb1 = 2.f * r1 - r21;
        if (ri >= 1 && ri <= TH && rj >= 1 && rj <= TW) {
          float2 ro; ro.x = r20 + RHO_F * (r0 - r20); ro.y = r21 + RHO_F * (r1 - r21);
          *(float2*)(r2_out + (size_t)gidx * 2) = ro;
        }
      }
      rbs[t * 2 + 0] = rb0;
      rbs[t * 2 + 1] = rb1;
    }
    __syncthreads();

    // ---- stage 3: u-update on the tile interior
    for (int t = tid; t < TH * TW; t += NTHREADS) {
      const int rh = t / TW;
      const int rw = t - rh * TW;
      const int gi = h0 + rh;
      const int gj = w0 + rw;
      const int xr = rh + 1;  // p in xb/rb coords
      const int xc = rw + 1;
      #define XB(rr, cc)    xbs[((rr) * XBC) + (cc)]
      #define RB(k, rr, cc) rbs[(((rr) * RBC) + (cc)) * 2 + (k)]
      // D = nabla2(xb) - rb, evaluated at (gi+di, gj+dj)
      #define D0(di, dj) ( (((gi + (di)) < HDIM - 1) ? (XB(xr + (di) + 1, xc + (dj)) - XB(xr + (di), xc + (dj))) : 0.f) \
                           - RB(0, xr + (di), xc + (dj)) )
      #define D1(di, dj) ( (((gj + (dj)) < WDIM - 1) ? (XB(xr + (di), xc + (dj) + 1) - XB(xr + (di), xc + (dj))) : 0.f) \
                           - RB(1, xr + (di), xc + (dj)) )
      const float d0c = D0(0, 0);
      const float d1c = D1(0, 0);
      const float G0 = d0c - ((gi > 0) ? D0(-1, 0) : 0.f);
      const float G1 = (gj > 0) ? (d0c - D0(0, -1)) : 0.f;
      const float G2 = d1c - ((gj > 0) ? D1(0, -1) : 0.f);
      const float G3 = (gi < HDIM - 1) ? (D1(1, 0) - d1c) : 0.f;

      float uo0 = 0.f, uo1 = 0.f, uo2 = 0.f, uo3 = 0.f;
      if (!first) {
        const float* cell = &u2s[(((rh + 3) * U2C) + (rw + 3)) * 4];
        uo0 = cell[0]; uo1 = cell[1]; uo2 = cell[2]; uo3 = cell[3];
      }
      const float uu0 = uo0 + SIGMA_F * G0;
      const float uu1 = uo1 + SIGMA_F * G1;
      const float uu2 = uo2 + SIGMA_F * G2;
      const float uu3 = uo3 + SIGMA_F * G3;
      const float nrm = sqrtf(uu0 * uu0 + uu1 * uu1 + uu2 * uu2 + uu3 * uu3);
      const float den = fmaxf(nrm * inv_lam2, 1.0f);
      const float inv_den = 1.0f / den;
      const float un0 = uu0 * inv_den;
      const float un1 = uu1 * inv_den;
      const float un2 = uu2 * inv_den;
      const float un3 = uu3 * inv_den;
      const int gidx = bbase + gi * WDIM + gj;
      float4 o;
      o.x = uo0 + RHO_F * (un0 - uo0);
      o.y = uo1 + RHO_F * (un1 - uo1);
      o.z = uo2 + RHO_F * (un2 - uo2);
      o.w = uo3 + RHO_F * (un3 - uo3);
      *(float4*)(u2_out + (size_t)gidx * 4) = o;
    }
  }
}

extern "C" void kernel_launch(void* const* d_in, const int* in_sizes, int n_in,
                              void* d_out, int out_size, void* d_ws, size_t ws_size,
                              hipStream_t stream) {
  (void)in_sizes; (void)n_in; (void)out_size; (void)ws_size;
  const float* y   = (const float*)d_in[0];
  const float* ths = (const float*)d_in[1];
  float* out = (float*)d_out;
  const size_t N = (size_t)NB * HDIM * WDIM;   // 4 Mi pixels

  // Workspace layout (13*N floats = 208 MB): ping-pong state buffers.
  float* ws  = (float*)d_ws;
  float* x2a = ws;              // N
  float* r2a = x2a + N;         // 2N
  float* r2b = r2a + 2 * N;     // 2N
  float* u2a = r2b + 2 * N;     // 4N
  float* u2b = u2a + 4 * N;     // 4N
  float* x2b = out;             // d_out doubles as the odd-parity x2 buffer

  dim3 grid(WDIM / TW, HDIM / TH, NB);  // (16, 64, 4)
  dim3 block(NTHREADS);
  const int NIT = 20;
  for (int it = 0; it < NIT; ++it) {
    const int first = (it == 0);
    const int last  = (it == NIT - 1);
    const int even  = ((it & 1) == 0);
    const float* x2i = first ? y : (even ? x2b : x2a);  // it=0 reads x2 = y
    const float* r2i = even ? r2b : r2a;                // never read on it=0 (first flag)
    const float* u2i = even ? u2b : u2a;
    float* x2o = even ? x2a : x2b;                      // it=19 (odd) -> d_out
    float* r2o = even ? r2a : r2b;
    float* u2o = even ? u2a : u2b;
    tgv_pdhg_step<<<grid, block, 0, stream>>>(x2i, r2i, u2i, y,
                                              x2o, r2o, u2o, ths, first, last);
  }
}